// MTH_with_neg_18622978195603
// MI455X (gfx1250) — compile-verified
//
#include <hip/hip_runtime.h>
#include <stdint.h>

typedef float v4f __attribute__((ext_vector_type(4)));

#define TSTEPS 4
#define BLOCK  256
#define CHUNK_PIX (BLOCK * 4)   // 1024 pixels (each thread: one float4) per chunk

// Largest thresh*2^-k (k in [0,7]) such that 2^-k <= r ; 0 if none (or r<=0/NaN).
// Exact power-of-2 comparisons => implement via exponent field + v_ldexp_f32.
__device__ __forceinline__ float quant_pow2_k8(float r, float th) {
  int eb = (__float_as_int(r) >> 23) & 0xff;   // biased exponent of |r|
  eb = eb > 127 ? 127 : eb;                    // clamp k >= 0  (2^-k <= 1)
  float v = ldexpf(th, eb - 127);              // thresh * 2^floor(log2 r), clamped
  return (r >= 0.0078125f) ? v : 0.0f;         // require 2^-7 <= r, kill r<=0/NaN
}

// One scan step for one scalar lane-element. Both pos_k and neg_k are monotone
// in k, so the K-sum of thre_k * diff_k(...) telescopes to a pow2 quantizer.
__device__ __forceinline__ float mth_step(float xv, float th, float inv075,
                                          float invth, float& mem, float& cum) {
  mem += xv;
  float sp = quant_pow2_k8(mem * inv075, th);                 // mem >= 0.75*thre_k
  float rn = fminf(-mem * inv075, cum * invth);               // AND of two monotone conds
  float sn = quant_pow2_k8(rn, th);
  float spike = sp - sn;
  mem -= spike;
  cum += spike;
  return spike;
}

__global__ __launch_bounds__(BLOCK) void mth_async_kernel(
    const float* __restrict__ x, const float* __restrict__ thresh,
    float* __restrict__ out, unsigned npix, unsigned nchunks) {
  // [2 buffers][T][threads] float4 : 32 KB. Each thread reads back exactly the
  // LDS slot its own async load wrote -> per-wave s_wait_asynccnt, no barriers.
  __shared__ v4f tile[2][TSTEPS][BLOCK];
  const unsigned tid = threadIdx.x;

  const float th     = thresh[0];
  const float inv075 = 1.0f / (0.75f * th);
  const float invth  = 1.0f / th;

  unsigned c   = blockIdx.x;
  int      buf = 0;

  if (c < nchunks) {  // prologue: async-prefetch first chunk (T planes)
    const unsigned base = c * CHUNK_PIX + tid * 4u;
#pragma unroll
    for (int t = 0; t < TSTEPS; ++t) {
      uint64_t ga = (uint64_t)(uintptr_t)(x + (size_t)t * npix + base);
      uint32_t la = (uint32_t)(uintptr_t)&tile[0][t][tid];
      asm volatile("global_load_async_to_lds_b128 %0, %1, off"
                   :: "v"(la), "v"(ga) : "memory");
    }
  }

  while (c < nchunks) {
    const unsigned cn = c + gridDim.x;          // block-uniform control flow
    if (cn < nchunks) {                          // prefetch next chunk into other buffer
      const unsigned nbase = cn * CHUNK_PIX + tid * 4u;
#pragma unroll
      for (int t = 0; t < TSTEPS; ++t) {
        uint64_t ga = (uint64_t)(uintptr_t)(x + (size_t)t * npix + nbase);
        uint32_t la = (uint32_t)(uintptr_t)&tile[buf ^ 1][t][tid];
        asm volatile("global_load_async_to_lds_b128 %0, %1, off"
                     :: "v"(la), "v"(ga) : "memory");
      }
      // 8 async loads outstanding; async loads retire in order -> <=4 means
      // the current chunk's 4 loads have landed in LDS.
      asm volatile("s_wait_asynccnt 4" ::: "memory");
    } else {
      asm volatile("s_wait_asynccnt 0" ::: "memory");
    }

    // ---- compute current chunk (4 pixels/thread, scan over T in registers) ----
    const unsigned base = c * CHUNK_PIX + tid * 4u;
    float mem[4] = {0.f, 0.f, 0.f, 0.f};
    float cum[4] = {0.f, 0.f, 0.f, 0.f};
#pragma unroll
    for (int t = 0; t < TSTEPS; ++t) {
      v4f xv = tile[buf][t][tid];               // ds_load_b128
      v4f sv;
#pragma unroll
      for (int i = 0; i < 4; ++i)
        sv[i] = mth_step(xv[i], th, inv075, invth, mem[i], cum[i]);
      // streamed once, never re-read: non-temporal 128-bit store
      __builtin_nontemporal_store(sv, (v4f*)(out + (size_t)t * npix + base));
    }

    buf ^= 1;
    c = cn;
  }
}

// Generic scalar fallback for any pixels not covered by full chunks.
__global__ void mth_tail_kernel(const float* __restrict__ x,
                                const float* __restrict__ thresh,
                                float* __restrict__ out,
                                unsigned npix, unsigned start) {
  unsigned p = start + blockIdx.x * blockDim.x + threadIdx.x;
  if (p >= npix) return;
  const float th     = thresh[0];
  const float inv075 = 1.0f / (0.75f * th);
  const float invth  = 1.0f / th;
  float mem = 0.f, cum = 0.f;
  for (int t = 0; t < TSTEPS; ++t) {
    float xv = x[(size_t)t * npix + p];
    out[(size_t)t * npix + p] = mth_step(xv, th, inv075, invth, mem, cum);
  }
}

extern "C" void kernel_launch(void* const* d_in, const int* in_sizes, int n_in,
                              void* d_out, int out_size, void* d_ws, size_t ws_size,
                              hipStream_t stream) {
  (void)n_in; (void)out_size; (void)d_ws; (void)ws_size;
  const float* x      = (const float*)d_in[0];
  const float* thresh = (const float*)d_in[1];
  float*       out    = (float*)d_out;

  const unsigned total = (unsigned)in_sizes[0];
  const unsigned npix  = total / TSTEPS;       // B*C*H*W = 8,388,608 here

  // float4/B128 path requires 16B-aligned time planes (npix % 4 == 0).
  const unsigned nchunks = ((npix % 4u) == 0u) ? (npix / CHUNK_PIX) : 0u;

  if (nchunks > 0) {
    unsigned grid = (nchunks + 3u) / 4u;       // ~4 chunks/block => 2048 blocks
    mth_async_kernel<<<dim3(grid), dim3(BLOCK), 0, stream>>>(x, thresh, out,
                                                             npix, nchunks);
  }
  const unsigned start = nchunks * CHUNK_PIX;
  const unsigned rem   = npix - start;
  if (rem > 0) {
    mth_tail_kernel<<<dim3((rem + 255u) / 256u), dim3(256), 0, stream>>>(
        x, thresh, out, npix, start);
  }
}